// SimpleGCNet_10926396801128
// MI455X (gfx1250) — compile-verified
//
#include <hip/hip_runtime.h>
#include <hip/hip_bf16.h>
#include <stdint.h>

#define NEG_SLOPE 0.01f

typedef float v2f __attribute__((ext_vector_type(2)));
typedef float v8f __attribute__((ext_vector_type(8)));

// ---------------------------------------------------------------------------
// Degree / normalization kernels
// ---------------------------------------------------------------------------
__global__ void k_deg_init(float* __restrict__ deg, int N) {
  int i = blockIdx.x * blockDim.x + threadIdx.x;
  if (i < N) deg[i] = 1.0f;  // self-loop contributes 1 to every node's degree
}

__global__ void k_deg_accum(const long long* __restrict__ dst,
                            float* __restrict__ deg, int E) {
  int e = blockIdx.x * blockDim.x + threadIdx.x;
  if (e < E) atomicAdd(&deg[(int)dst[e]], 1.0f);
}

__global__ void k_dinv(float* __restrict__ deg, int N) {
  int i = blockIdx.x * blockDim.x + threadIdx.x;
  if (i < N) {
    float d = deg[i];
    deg[i] = (d > 0.0f) ? rsqrtf(d) : 0.0f;  // in-place deg -> dinv
  }
}

// ---------------------------------------------------------------------------
// fp32 GEMM via V_WMMA_F32_16X16X4_F32: one 16x16 output tile per wave.
// A (16x4 f32): lanes 0-15 hold K=k0,k0+1 in v[0],v[1]; lanes 16-31 hold K=k0+2,k0+3.
// B (4x16 f32): lane = n + 16*(k>=2), vgpr = k&1.
// C/D (16x16 f32): VGPR r, lane l -> M = r + 8*(l>>4), N = l&15.
// ---------------------------------------------------------------------------
__global__ void k_gemm_wmma(const float* __restrict__ X, const float* __restrict__ W,
                            float* __restrict__ Y, int N, int K, int F) {
  int gtid  = blockIdx.x * blockDim.x + threadIdx.x;
  int wave  = gtid >> 5;
  int lane  = threadIdx.x & 31;
  int tilesN = F >> 4;
  int tm = wave / tilesN;
  int tn = wave - tm * tilesN;
  int m0 = tm << 4;
  int n0 = tn << 4;
  if (m0 >= N) return;

  int half = lane >> 4;   // 0: K pair {0,1}; 1: K pair {2,3}
  int l16  = lane & 15;

  if (m0 + 16 <= N) {
    v8f acc = {};
    const float* arow = X + (size_t)(m0 + l16) * K + 2 * half;        // A: M=l16
    const float* bcol = W + (size_t)(2 * half) * F + n0 + l16;        // B: N=l16
    for (int k0 = 0; k0 < K; k0 += 4) {
      v2f a, b;
      a.x = arow[k0];
      a.y = arow[k0 + 1];
      b.x = bcol[(size_t)k0 * F];
      b.y = bcol[(size_t)(k0 + 1) * F];
      acc = __builtin_amdgcn_wmma_f32_16x16x4_f32(
          /*neg_a=*/false, a, /*neg_b=*/false, b,
          /*c_mod=*/(short)0, acc, /*reuse_a=*/false, /*reuse_b=*/false);
    }
    float* yrow = Y + (size_t)(m0 + 8 * half) * F + n0 + l16;
#pragma unroll
    for (int r = 0; r < 8; ++r) yrow[(size_t)r * F] = acc[r];
  } else {
    // remainder tile (not hit for N=100000): scalar fallback, wave-uniform branch
    int m = m0 + lane;
    if (lane < 16 && m < N) {
      for (int n = 0; n < 16; ++n) {
        float s = 0.0f;
        for (int k = 0; k < K; ++k) s = fmaf(X[(size_t)m * K + k], W[(size_t)k * F + n0 + n], s);
        Y[(size_t)m * F + n0 + n] = s;
      }
    }
  }
}

// ---------------------------------------------------------------------------
// out[i,f] = b[f] + dinv[i]^2 * xw[i,f]   (bias + self-loop term, no atomics)
// ---------------------------------------------------------------------------
__global__ void k_self_bias(const float* __restrict__ xw, const float* __restrict__ dinv,
                            const float* __restrict__ b, float* __restrict__ out,
                            int N, int F) {
  int t = blockIdx.x * blockDim.x + threadIdx.x;
  if (t >= N * F) return;
  int i = t / F;
  int f = t - i * F;
  float di = dinv[i];
  out[t] = b[f] + di * di * xw[t];
}

// ---------------------------------------------------------------------------
// Edge scatter: out[dst] += xw[src] * dinv[src]*dinv[dst]
// One thread per (edge, 4-feature group); float4 gather + 4 float atomics.
// All hot arrays fit in the 192 MB L2, so these resolve at L2 bandwidth.
// ---------------------------------------------------------------------------
__global__ void k_agg_edges(const long long* __restrict__ src,
                            const long long* __restrict__ dst,
                            const float* __restrict__ dinv,
                            const float* __restrict__ xw,
                            float* __restrict__ out,
                            int E, int F) {
  int groups = F >> 2;
  long long t = (long long)blockIdx.x * blockDim.x + threadIdx.x;
  if (t >= (long long)E * groups) return;
  int e = (int)(t / groups);
  int g = (int)(t - (long long)e * groups);
  int s = (int)src[e];
  int d = (int)dst[e];
  float nrm = dinv[s] * dinv[d];
  const float4 v = *(const float4*)(xw + (size_t)s * F + g * 4);
  float* o = out + (size_t)d * F + g * 4;
  atomicAdd(o + 0, v.x * nrm);
  atomicAdd(o + 1, v.y * nrm);
  atomicAdd(o + 2, v.z * nrm);
  atomicAdd(o + 3, v.w * nrm);
}

__global__ void k_leaky(float* __restrict__ h, long long n) {
  long long t = (long long)blockIdx.x * blockDim.x + threadIdx.x;
  if (t < n) {
    float v = h[t];
    h[t] = (v >= 0.0f) ? v : NEG_SLOPE * v;
  }
}

// ---------------------------------------------------------------------------
// Final projection: [N,32] @ [32,10] + b ; tiny, plain VALU.
// ---------------------------------------------------------------------------
__global__ void k_proj(const float* __restrict__ h, const float* __restrict__ wp,
                       const float* __restrict__ bp, float* __restrict__ out, int N) {
  int t = blockIdx.x * blockDim.x + threadIdx.x;
  if (t >= N * 10) return;
  int n = t / 10;
  int c = t - n * 10;
  const float* hr = h + (size_t)n * 32;
  float s = bp[c];
#pragma unroll
  for (int k = 0; k < 32; ++k) s = fmaf(hr[k], wp[k * 10 + c], s);
  out[t] = s;
}

// ---------------------------------------------------------------------------
static inline int cdivll(long long a, long long b) { return (int)((a + b - 1) / b); }

extern "C" void kernel_launch(void* const* d_in, const int* in_sizes, int n_in,
                              void* d_out, int out_size, void* d_ws, size_t ws_size,
                              hipStream_t stream) {
  const int F_IN = 128;
  const int N = in_sizes[0] / F_IN;       // 100000
  const int E = in_sizes[1] / 2;          // 3200000

  const float*     x   = (const float*)d_in[0];
  const long long* ei  = (const long long*)d_in[1];  // int64 [2, E]
  const long long* src = ei;
  const long long* dst = ei + E;
  const float* w1 = (const float*)d_in[2]; const float* b1 = (const float*)d_in[3];
  const float* w2 = (const float*)d_in[4]; const float* b2 = (const float*)d_in[5];
  const float* w3 = (const float*)d_in[6]; const float* b3 = (const float*)d_in[7];
  const float* wp = (const float*)d_in[8]; const float* bp = (const float*)d_in[9];

  // Workspace layout (floats): dinv[N] | xw[N*64] | hA[N*64] | hB[N*64]
  float* ws   = (float*)d_ws;
  float* dinv = ws;
  float* xw   = dinv + (size_t)N;
  float* hA   = xw + (size_t)N * 64;
  float* hB   = hA + (size_t)N * 64;

  const int B = 256;

  // --- degree normalization (shared by all three conv layers) ---
  k_deg_init <<<cdivll(N, B), B, 0, stream>>>(dinv, N);
  k_deg_accum<<<cdivll(E, B), B, 0, stream>>>(dst, dinv, E);
  k_dinv     <<<cdivll(N, B), B, 0, stream>>>(dinv, N);

  // --- one GCN layer ---
  auto layer = [&](const float* in, int Fin, int Fout,
                   const float* w, const float* b, float* outbuf) {
    long long tiles = (long long)((N + 15) / 16) * (Fout / 16);
    k_gemm_wmma<<<cdivll(tiles * 32, B), B, 0, stream>>>(in, w, xw, N, Fin, Fout);
    k_self_bias<<<cdivll((long long)N * Fout, B), B, 0, stream>>>(xw, dinv, b, outbuf, N, Fout);
    k_agg_edges<<<cdivll((long long)E * (Fout / 4), B), B, 0, stream>>>(src, dst, dinv, xw, outbuf, E, Fout);
    k_leaky    <<<cdivll((long long)N * Fout, B), B, 0, stream>>>(outbuf, (long long)N * Fout);
  };

  layer(x,  128, 32, w1, b1, hA);
  layer(hA,  32, 64, w2, b2, hB);
  layer(hB,  64, 32, w3, b3, hA);

  k_proj<<<cdivll((long long)N * 10, B), B, 0, stream>>>(hA, wp, bp, (float*)d_out, N);
}